// S2Convolution_39213051412535
// MI455X (gfx1250) — compile-verified
//
#include <hip/hip_runtime.h>
#include <hip/hip_bf16.h>

// CDNA5 fp32 WMMA operand types: 16x16x4_f32 -> A,B = 2 VGPRs (v2f), C/D = 8 VGPRs (v8f)
typedef __attribute__((ext_vector_type(2))) float v2f;
typedef __attribute__((ext_vector_type(8))) float v8f;

// Problem constants
// B=1024, TI=2, TE=3 (ab=6), B_IN=16, B_OUT=32, DEGREES={0,2,4,6,8}

// ---------------------------------------------------------------------------
// Kernel A: degree l=0 (nsh=1).  out0[n,ab,o] = sum_i x0[n,ab,i] * w0[ab,i,o] + bias[ab,o]
// Tile: M=n (16), N=o (16), K=i (16 = 4 x wmma K=4). Fully coalesced stores.
// x0 strides: n=96, ab=16, i=1.  w0 strides: ab=512, i=32, o=1.
// out0 strides: n=192, ab=32, o=1. bias: ab*32+o.
// ---------------------------------------------------------------------------
__global__ __launch_bounds__(256) void s2conv_l0_kernel(
    const float* __restrict__ x0, const float* __restrict__ w0,
    const float* __restrict__ bias, float* __restrict__ out0) {
  int wid  = blockIdx.x * 8 + (threadIdx.x >> 5);   // 768 waves total
  int lane = threadIdx.x & 31;
  int hi = lane >> 4;      // half-wave selector
  int m  = lane & 15;      // A: row (n offset); B/D: column (o offset)

  int ot = wid & 1;  int t = wid >> 1;
  int ab = t % 6;    int nt = t / 6;
  int n0 = nt * 16;
  int o0 = ot * 16;

  // ---- load A (x tile): lane -> row M=m, K pairs per wmma step
  float av[8];
  int xbase = (n0 + m) * 96 + ab * 16;
#pragma unroll
  for (int kk = 0; kk < 4; ++kk) {
    int i0 = 4 * kk + 2 * hi;
    av[2 * kk + 0] = x0[xbase + i0];
    av[2 * kk + 1] = x0[xbase + i0 + 1];
  }
  // ---- load B (w tile): lane -> col N=m
  float bv[8];
  int wbase = ab * 512 + o0 + m;
#pragma unroll
  for (int kk = 0; kk < 4; ++kk) {
    int i0 = 4 * kk + 2 * hi;
    bv[2 * kk + 0] = w0[wbase + (i0 + 0) * 32];
    bv[2 * kk + 1] = w0[wbase + (i0 + 1) * 32];
  }

  v8f c = {};
#pragma unroll
  for (int kk = 0; kk < 4; ++kk) {
    v2f a = {av[2 * kk], av[2 * kk + 1]};
    v2f b = {bv[2 * kk], bv[2 * kk + 1]};
    c = __builtin_amdgcn_wmma_f32_16x16x4_f32(false, a, false, b, (short)0, c,
                                              false, false);
  }

  float bs = bias[ab * 32 + o0 + m];   // broadcast over n rows
  int ob = ab * 32 + o0 + m;
#pragma unroll
  for (int r = 0; r < 8; ++r) {
    int row = n0 + r + 8 * hi;         // C/D layout: VGPR r -> M = r (+8 for hi lanes)
    out0[row * 192 + ob] = c[r] + bs;  // o contiguous across lanes -> 64B runs
  }
}

// ---------------------------------------------------------------------------
// Kernel B: degrees l=2,4,6,8.  For fixed (n,ab,o):
//   out[li,k] = sum_i x[n,ab,i,li] * w[ab,i,o,k]   (nsh x nsh, k contiguous)
// Tile: M=li, N=k, K=i.  A tile held in registers, reused over 8 o values.
// Edge masking on loads/stores only; EXEC all-1 at the WMMAs.
// x strides (deg l): n=96*nsh, ab=16*nsh, i=nsh, li=1
// w strides: ab=512*nsh, i=32*nsh, o=nsh, k=1
// out strides: n=192*nsh2, ab=32*nsh2, o=nsh2, li=nsh, k=1
// ---------------------------------------------------------------------------
__global__ __launch_bounds__(256) void s2conv_hi_kernel(
    const float* __restrict__ x2, const float* __restrict__ x4,
    const float* __restrict__ x6, const float* __restrict__ x8,
    const float* __restrict__ w2, const float* __restrict__ w4,
    const float* __restrict__ w6, const float* __restrict__ w8,
    float* __restrict__ out2, float* __restrict__ out4,
    float* __restrict__ out6, float* __restrict__ out8) {
  int wid  = blockIdx.x * 8 + (threadIdx.x >> 5);   // 1024*6*7*4 = 172032 waves
  int lane = threadIdx.x & 31;
  int hi = lane >> 4;
  int m  = lane & 15;

  int oc = wid & 3;  int t = wid >> 2;   // o chunk (8 outputs each)
  int ts = t % 7;    t /= 7;             // tile slot: 0..2 -> l=2,4,6 ; 3..6 -> l=8 quadrants
  int ab = t % 6;
  int n  = t / 6;

  const float* __restrict__ x;
  const float* __restrict__ w;
  float* __restrict__ out;
  int nsh, li0 = 0, k0 = 0;
  if (ts == 0)      { x = x2; w = w2; out = out2; nsh = 5; }
  else if (ts == 1) { x = x4; w = w4; out = out4; nsh = 9; }
  else if (ts == 2) { x = x6; w = w6; out = out6; nsh = 13; }
  else {
    x = x8; w = w8; out = out8; nsh = 17;
    li0 = ((ts - 3) >> 1) * 16;
    k0  = ((ts - 3) & 1) * 16;
  }

  // ---- A tile: X[li=m][i], masked for li >= nsh (edge tiles of l=8)
  int li  = li0 + m;
  bool arow = (li < nsh);
  int xbase = n * 96 * nsh + ab * 16 * nsh + li;
  float av[8];
#pragma unroll
  for (int kk = 0; kk < 4; ++kk) {
    int i0 = 4 * kk + 2 * hi;
    av[2 * kk + 0] = arow ? x[xbase + (i0 + 0) * nsh] : 0.0f;
    av[2 * kk + 1] = arow ? x[xbase + (i0 + 1) * nsh] : 0.0f;
  }

  int k = k0 + m;
  bool kcol = (k < nsh);
  int nsh2 = nsh * nsh;
  int wab = ab * 512 * nsh;
  int obase = n * 192 * nsh2 + ab * 32 * nsh2 + k;

  for (int oo = 0; oo < 8; ++oo) {
    int o = oc * 8 + oo;
    // ---- B tile: Wt[i][k=m] for this o, masked for k >= nsh
    int wbase = wab + o * nsh + k;
    float bv[8];
#pragma unroll
    for (int kk = 0; kk < 4; ++kk) {
      int i0 = 4 * kk + 2 * hi;
      bv[2 * kk + 0] = kcol ? w[wbase + (i0 + 0) * 32 * nsh] : 0.0f;
      bv[2 * kk + 1] = kcol ? w[wbase + (i0 + 1) * 32 * nsh] : 0.0f;
    }

    v8f c = {};
#pragma unroll
    for (int kk = 0; kk < 4; ++kk) {
      v2f a = {av[2 * kk], av[2 * kk + 1]};
      v2f b = {bv[2 * kk], bv[2 * kk + 1]};
      c = __builtin_amdgcn_wmma_f32_16x16x4_f32(false, a, false, b, (short)0, c,
                                                false, false);
    }

    // ---- store: lane -> k=k0+m (contiguous across lanes), VGPR r -> li
    int ob = obase + o * nsh2;
#pragma unroll
    for (int r = 0; r < 8; ++r) {
      int lir = li0 + r + 8 * hi;
      if (kcol && (lir < nsh)) out[ob + lir * nsh] = c[r];
    }
  }
}

extern "C" void kernel_launch(void* const* d_in, const int* in_sizes, int n_in,
                              void* d_out, int out_size, void* d_ws, size_t ws_size,
                              hipStream_t stream) {
  const float* x0 = (const float*)d_in[0];
  const float* w0 = (const float*)d_in[1];
  const float* x2 = (const float*)d_in[2];
  const float* w2 = (const float*)d_in[3];
  const float* x4 = (const float*)d_in[4];
  const float* w4 = (const float*)d_in[5];
  const float* x6 = (const float*)d_in[6];
  const float* w6 = (const float*)d_in[7];
  const float* x8 = (const float*)d_in[8];
  const float* w8 = (const float*)d_in[9];
  const float* bias = (const float*)d_in[10];

  float* out  = (float*)d_out;
  const size_t base = 1024u * 2u * 3u * 32u;  // 196608 elements per nsh^2 unit
  float* out0 = out;
  float* out2 = out0 + base * 1;    // nsh^2 = 1
  float* out4 = out2 + base * 25;   // nsh^2 = 25
  float* out6 = out4 + base * 81;   // nsh^2 = 81
  float* out8 = out6 + base * 169;  // nsh^2 = 169

  // l=0: 64 n-tiles * 2 o-tiles * 6 ab = 768 waves -> 96 blocks of 8 waves
  s2conv_l0_kernel<<<96, 256, 0, stream>>>(x0, w0, bias, out0);

  // l=2,4,6,8: 1024 * 6 * 7 tile-slots * 4 o-chunks = 172032 waves -> 21504 blocks
  s2conv_hi_kernel<<<21504, 256, 0, stream>>>(x2, x4, x6, x8, w2, w4, w6, w8,
                                              out2, out4, out6, out8);
}